// GeodesicDistanceWeightedLoss_59270548684945
// MI455X (gfx1250) — compile-verified
//
#include <hip/hip_runtime.h>

// GeodesicDistanceWeightedLoss: mean(| (sum_{j<4} in[4g+j]*tg[4g+j])^2 - 1 |)
// Bandwidth-bound: 268 MB @ 23.3 TB/s => ~11.5us floor. Strategy:
//   - b128 non-temporal loads (stream > 192MB L2, touched once)
//   - x4-unrolled streaming loop: 8 b128 loads in flight per lane
//   - wave32 reduction via V_WMMA_F32_16X16X4_F32 (ones-matrix row-sum trick)
//   - deterministic two-pass block-partial reduction through d_ws

typedef __attribute__((ext_vector_type(4))) float v4f;
typedef __attribute__((ext_vector_type(2))) float v2f;
typedef __attribute__((ext_vector_type(8))) float v8f;

__device__ __forceinline__ float dot4(v4f a, v4f b) {
  float s = a[0] * b[0];
  s = __builtin_fmaf(a[1], b[1], s);
  s = __builtin_fmaf(a[2], b[2], s);
  s = __builtin_fmaf(a[3], b[3], s);
  return s;
}

__device__ __forceinline__ float sq_abs_err(float s) {
  return __builtin_fabsf(__builtin_fmaf(s, s, -1.0f));
}

__device__ __forceinline__ float wave32_sum(float acc) {
#if __has_builtin(__builtin_amdgcn_wmma_f32_16x16x4_f32)
  // A (16x4 f32): VGPR0 = lanes 0-15 -> (M=lane,K=0), lanes 16-31 -> (M=lane-16,K=2).
  // With a.y = 0 and B = ones(4x16): D[m][n] = acc[m] + acc[m+16]  (rowsum, all n).
  // Lane L<16 holds D rows 0..7 in its 8 dwords; lane L>=16 holds rows 8..15.
  v2f a; a[0] = acc; a[1] = 0.0f;
  v2f b; b[0] = 1.0f; b[1] = 1.0f;
  v8f c = {};
  v8f d = __builtin_amdgcn_wmma_f32_16x16x4_f32(
      /*neg_a=*/false, a, /*neg_b=*/false, b,
      /*c_mod=*/(short)0, c, /*reuse_a=*/false, /*reuse_b=*/false);
  float p = d[0] + d[1] + d[2] + d[3] + d[4] + d[5] + d[6] + d[7];
  p += __shfl_down(p, 16, 32);   // combine rows 0..7 with rows 8..15
  return p;                       // lane 0 holds the wave total
#else
  for (int off = 16; off > 0; off >>= 1) acc += __shfl_down(acc, off, 32);
  return acc;
#endif
}

__device__ __forceinline__ float block256_sum(float acc, float* smem) {
  float w = wave32_sum(acc);
  const int lane = threadIdx.x & 31;
  const int wave = threadIdx.x >> 5;
  if (lane == 0) smem[wave] = w;
  __syncthreads();
  float total = 0.0f;
  if (threadIdx.x == 0) {
    const int nw = blockDim.x >> 5;
    for (int i = 0; i < nw; ++i) total += smem[i];
  }
  return total;  // valid in thread 0 only
}

__global__ void geo_partial_kernel(const float* __restrict__ in,
                                   const float* __restrict__ tg,
                                   float* __restrict__ partials,
                                   int num_groups, int main_iters) {
  __shared__ float smem[8];
  const v4f* __restrict__ a4 = (const v4f*)in;
  const v4f* __restrict__ b4 = (const v4f*)tg;
  const int stride = gridDim.x * blockDim.x;
  int g = blockIdx.x * blockDim.x + threadIdx.x;
  float acc = 0.0f;

  // Main stage: uniform trip count, unrolled x4 -> 8 b128 NT loads issued
  // back-to-back per iteration before any s_wait_loadcnt.
  for (int i = 0; i < main_iters; i += 4) {
    v4f a0 = __builtin_nontemporal_load(a4 + g);
    v4f b0 = __builtin_nontemporal_load(b4 + g);
    v4f a1 = __builtin_nontemporal_load(a4 + g + stride);
    v4f b1 = __builtin_nontemporal_load(b4 + g + stride);
    v4f a2 = __builtin_nontemporal_load(a4 + g + 2 * stride);
    v4f b2 = __builtin_nontemporal_load(b4 + g + 2 * stride);
    v4f a3 = __builtin_nontemporal_load(a4 + g + 3 * stride);
    v4f b3 = __builtin_nontemporal_load(b4 + g + 3 * stride);
    g += 4 * stride;
    acc += sq_abs_err(dot4(a0, b0));
    acc += sq_abs_err(dot4(a1, b1));
    acc += sq_abs_err(dot4(a2, b2));
    acc += sq_abs_err(dot4(a3, b3));
  }
  // Tail: bound-checked grid-stride (empty for the reference shape).
  for (; g < num_groups; g += stride) {
    v4f a = __builtin_nontemporal_load(a4 + g);
    v4f b = __builtin_nontemporal_load(b4 + g);
    acc += sq_abs_err(dot4(a, b));
  }

  float t = block256_sum(acc, smem);
  if (threadIdx.x == 0) partials[blockIdx.x] = t;
}

__global__ void geo_finalize_kernel(const float* __restrict__ partials,
                                    int n, float* __restrict__ out,
                                    float inv_groups) {
  __shared__ float smem[8];
  float acc = 0.0f;
  for (int i = threadIdx.x; i < n; i += blockDim.x) acc += partials[i];
  float t = block256_sum(acc, smem);
  if (threadIdx.x == 0) out[0] = t * inv_groups;
}

extern "C" void kernel_launch(void* const* d_in, const int* in_sizes, int n_in,
                              void* d_out, int out_size, void* d_ws, size_t ws_size,
                              hipStream_t stream) {
  const float* in = (const float*)d_in[0];
  const float* tg = (const float*)d_in[1];
  float* out = (float*)d_out;
  float* partials = (float*)d_ws;

  const int N = in_sizes[0];
  const int num_groups = N / 4;  // 8,388,608 for the reference shape

  int blocks = 4096;  // 2^20 lanes -> exactly 8 groups/lane; saturates HBM MLP
  const int max_blocks = (int)(ws_size / sizeof(float));
  if (blocks > max_blocks) blocks = max_blocks > 0 ? max_blocks : 1;
  const int max_useful = (num_groups + 255) / 256;
  if (blocks > max_useful) blocks = max_useful;

  const int lanes = blocks * 256;
  const int iters = num_groups / lanes;   // full passes every lane performs
  const int main_iters = iters & ~3;      // multiple of 4 for the unrolled stage

  geo_partial_kernel<<<blocks, 256, 0, stream>>>(in, tg, partials, num_groups,
                                                 main_iters);
  geo_finalize_kernel<<<1, 256, 0, stream>>>(partials, blocks, out,
                                             1.0f / (float)num_groups);
}